// PrototypicalNetwork_57037165691753
// MI455X (gfx1250) — compile-verified
//
#include <hip/hip_runtime.h>
#include <math.h>

typedef __attribute__((ext_vector_type(2))) float v2f;
typedef __attribute__((ext_vector_type(8))) float v8f;

#define N_CLASSES 256
#define PER_CLASS 128
#define N_SUPPORT 5
#define N_QUERY   (PER_CLASS - N_SUPPORT)   // 123
#define N_VIEW    2
#define DIM       384
#define KPAIRS    (DIM / 2)                  // 192
#define LSTRIDE   261                        // 261 mod 64 = 5 -> conflict-free row scans

// workspace layout (in floats)
#define WS_PROTOP 0                                      // [2][192][256][2] K-pair-packed protos
#define WS_Y2     (N_VIEW * DIM * N_CLASSES)             // [256][2]  proto squared norms
#define WS_PART   (WS_Y2 + N_CLASSES * N_VIEW)           // [512]     per-block partial sums

// ---------------------------------------------------------------------------
// Kernel 1: prototypes (mean of 5 support rows), written K-pair-packed as
// protoP[v][k/2][m][k&1] so a WMMA B fragment (B[ka][n], B[ka+1][n]) is one
// contiguous 8-byte load per lane. Also y2[m][v].
// grid = 512 blocks (class*2+view), block = 128 threads.
// ---------------------------------------------------------------------------
__global__ void __launch_bounds__(128)
proto_kernel(const float* __restrict__ reps, float* __restrict__ ws) {
  const int c = blockIdx.x >> 1;
  const int v = blockIdx.x & 1;
  float* protoP = ws + WS_PROTOP;
  float* y2     = ws + WS_Y2;
  __shared__ float red[128];

  float ss = 0.0f;
  for (int d = threadIdx.x; d < DIM; d += 128) {
    float s = 0.0f;
    #pragma unroll
    for (int sp = 0; sp < N_SUPPORT; ++sp)
      s += reps[((size_t)((c * PER_CLASS + sp) * N_VIEW + v)) * DIM + d];
    s *= (1.0f / (float)N_SUPPORT);
    protoP[(((size_t)(v * KPAIRS + (d >> 1)) * N_CLASSES + c) << 1) + (d & 1)] = s;
    ss += s * s;
  }
  red[threadIdx.x] = ss;
  __syncthreads();
  for (int off = 64; off > 0; off >>= 1) {
    if ((int)threadIdx.x < off) red[threadIdx.x] += red[threadIdx.x + off];
    __syncthreads();
  }
  if (threadIdx.x == 0) y2[c * N_VIEW + v] = red[0];
}

// ---------------------------------------------------------------------------
// Kernel 2: per (class, view): xy = Q(128x384) * P^T(384x256) via
// V_WMMA_F32_16X16X4_F32, logits L = 2*xy - y2[m] (x2 cancels in softmax),
// then per-row log-sum-exp over 256 classes and correct-class extraction.
// grid = 512 blocks, block = 256 threads (8 waves x 16 query rows).
// Dynamic LDS: 128 * LSTRIDE floats.
// ---------------------------------------------------------------------------
__global__ void __launch_bounds__(256)
dist_loss_kernel(const float* __restrict__ reps, const float* __restrict__ ws,
                 float* __restrict__ partials) {
  extern __shared__ float slog[];            // [128][LSTRIDE] logits
  __shared__ float red[256];

  const float* protoP = ws + WS_PROTOP;
  const float* y2     = ws + WS_Y2;

  const int c    = blockIdx.x >> 1;
  const int v    = blockIdx.x & 1;
  const int lane = threadIdx.x & 31;
  const int wv   = threadIdx.x >> 5;         // wave id 0..7
  const int hi   = lane >> 4;                // half-wave select (K pair / M+8)
  const int ln   = lane & 15;

  const int rowTile = wv * 16;               // this wave's 16 query rows
  // A-fragment row for this lane (ISA: lanes 0-15 and 16-31 both hold M=0..15)
  const float* qptr =
      reps + ((size_t)((c * PER_CLASS + rowTile + ln) * N_VIEW + v)) * DIM;

  // Two N-halves of 128 classes each to bound accumulator VGPRs (8 x v8f).
  for (int half = 0; half < 2; ++half) {
    v8f acc[8] = {};
    for (int k0 = 0; k0 < DIM; k0 += 4) {
      const int ka = k0 + 2 * hi;            // A vgpr0/1 -> K = ka, ka+1 (ka even)
      const v2f a = *(const v2f*)(qptr + ka);
      // B fragment base: K-pair (ka, ka+1), N = half*128 + nt*16 + ln
      const float* bp =
          protoP + (((size_t)(v * KPAIRS + (ka >> 1)) * N_CLASSES
                     + half * 128 + ln) << 1);
      #pragma unroll
      for (int nt = 0; nt < 8; ++nt) {
        const v2f b = *(const v2f*)(bp + nt * 32);   // one b64 per fragment
        acc[nt] = __builtin_amdgcn_wmma_f32_16x16x4_f32(
            false, a, false, b, (short)0, acc[nt], false, false);
      }
    }
    // Logits to LDS. C/D layout: vgpr j holds M = j + 8*hi, N = ln.
    #pragma unroll
    for (int nt = 0; nt < 8; ++nt) {
      const int m   = half * 128 + nt * 16 + ln;
      const float y = y2[m * N_VIEW + v];
      #pragma unroll
      for (int j = 0; j < 8; ++j) {
        const int r = rowTile + 8 * hi + j;
        slog[r * LSTRIDE + m] = 2.0f * acc[nt][j] - y;
      }
    }
  }
  __syncthreads();

  // Per-row log-softmax; rows 0..4 are support rows (excluded).
  float contrib = 0.0f;
  if ((int)threadIdx.x >= N_SUPPORT && (int)threadIdx.x < PER_CLASS) {
    const float* rowp = slog + (size_t)threadIdx.x * LSTRIDE;
    float mx = -INFINITY;
    for (int m = 0; m < N_CLASSES; ++m) mx = fmaxf(mx, rowp[m]);
    float s = 0.0f;
    for (int m = 0; m < N_CLASSES; ++m) s += __expf(rowp[m] - mx);
    contrib = rowp[c] - (mx + __logf(s));    // log p(correct class)
  }
  red[threadIdx.x] = contrib;
  __syncthreads();
  for (int off = 128; off > 0; off >>= 1) {
    if ((int)threadIdx.x < off) red[threadIdx.x] += red[threadIdx.x + off];
    __syncthreads();
  }
  if (threadIdx.x == 0) partials[blockIdx.x] = red[0];
}

// ---------------------------------------------------------------------------
// Kernel 3: deterministic final reduction of 512 partials -> scalar loss.
// ---------------------------------------------------------------------------
__global__ void __launch_bounds__(512)
finalize_kernel(const float* __restrict__ partials, float* __restrict__ out) {
  __shared__ float red[512];
  red[threadIdx.x] = partials[threadIdx.x];
  __syncthreads();
  for (int off = 256; off > 0; off >>= 1) {
    if ((int)threadIdx.x < off) red[threadIdx.x] += red[threadIdx.x + off];
    __syncthreads();
  }
  if (threadIdx.x == 0)
    out[0] = -red[0] / (float)(N_CLASSES * N_QUERY * N_VIEW);
}

// ---------------------------------------------------------------------------
extern "C" void kernel_launch(void* const* d_in, const int* in_sizes, int n_in,
                              void* d_out, int out_size, void* d_ws, size_t ws_size,
                              hipStream_t stream) {
  const float* reps = (const float*)d_in[0];   // (32768, 2, 384) fp32
  // d_in[1] (labels) is implied by index order; unused.
  float* ws  = (float*)d_ws;
  float* out = (float*)d_out;

  proto_kernel<<<N_CLASSES * N_VIEW, 128, 0, stream>>>(reps, ws);

  const size_t lds_bytes = (size_t)PER_CLASS * LSTRIDE * sizeof(float); // ~131 KB (<320 KB/WGP)
  dist_loss_kernel<<<N_CLASSES * N_VIEW, 256, lds_bytes, stream>>>(
      reps, ws, ws + WS_PART);

  finalize_kernel<<<1, 512, 0, stream>>>(ws + WS_PART, out);
}